// VarConvND_34703335752142
// MI455X (gfx1250) — compile-verified
//
#include <hip/hip_runtime.h>

typedef float v2f __attribute__((ext_vector_type(2)));
typedef float v8f __attribute__((ext_vector_type(8)));

#define B_    64
#define CIN   32
#define HH    64
#define WW    64
#define COUT  16
#define KK    288     // CIN * 3 * 3
#define SS    4096    // 64 * 64
#define NS    8       // output locations per workgroup (one ow strip)
#define TW    10      // NS + 2 halo columns
#define BP    65      // batch dim padded to 65 dwords -> conflict-free LDS

// LDS tile: [CIN][3 rows][TW cols][BP batches] = 32*3*10*65 floats = 249600 B
#define TILE_FLOATS (CIN * 3 * TW * BP)

// LDS dword offset (excluding wv and batch) for unfold index k = c*9 + i*3 + j.
// Fully constant-folds after the K-loop is unrolled.
__device__ constexpr int lofs(int k) {
    const int c = k / 9, r = k % 9, i = r / 3, j = r % 3;
    return ((c * 3 + i) * TW + j) * BP;
}

__global__ __launch_bounds__(256) void varconv_wmma_f32(
    const float* __restrict__ X,     // [B, CIN, H, W]
    const float* __restrict__ Wt,    // [S, K, COUT]
    const float* __restrict__ bias,  // [H_OUT, W_OUT]
    float* __restrict__ out)         // [B, COUT, H_OUT, W_OUT]
{
    extern __shared__ float tile[];

    const int tid = threadIdx.x;
    const int bid = blockIdx.x;        // 512 blocks: 64 oh rows x 8 strips
    const int oh  = bid >> 3;
    const int ow0 = (bid & 7) << 3;

    // ---------------- stage X halo window into LDS ----------------
    for (int p = tid; p < B_ * CIN * 3; p += 256) {
        const int i  = p % 3;
        const int bc = p / 3;
        const int c  = bc % CIN;
        const int b  = bc / CIN;
        const int ih = oh + i - 1;
        float* dst = &tile[((c * 3 + i) * TW) * BP + b];
        if (ih < 0 || ih >= HH) {
#pragma unroll
            for (int t = 0; t < TW; ++t) dst[t * BP] = 0.0f;
        } else {
            const float* src = X + (((size_t)b * CIN + c) * HH + ih) * WW;
#pragma unroll
            for (int t = 0; t < TW; ++t) {
                const int iw = ow0 + t - 1;
                dst[t * BP] = (iw >= 0 && iw < WW) ? src[iw] : 0.0f;
            }
        }
    }
    __syncthreads();

    // ---------------- per-wave GEMM: one output location s ----------------
    const int wv   = tid >> 5;        // wave id = s offset within strip
    const int lane = tid & 31;
    const int half = lane >> 4;       // half-wave: K+0..1 vs K+2..3
    const int n    = lane & 15;       // N (c_out) / M-row (batch) lane index

    const int ow = ow0 + wv;
    const int s  = oh * WW + ow;
    const float bs = bias[s];

    // per-lane bases: all loop loads become immediate-offset accesses
    const float* wlane = Wt + (size_t)s * (KK * COUT) + half * (2 * COUT) + n;
    const float* tbase = tile + wv * BP + n;

    v8f acc0 = {}, acc1 = {}, acc2 = {}, acc3 = {};

#pragma unroll
    for (int k0 = 0; k0 < KK; k0 += 4) {
        if ((k0 & 15) == 0 && k0 + 16 < KK)      // weight stream ~1KB ahead
            __builtin_prefetch(wlane + (k0 + 16) * COUT, 0, 0);

        // constant after unroll; one v_cndmask each for the half-wave split
        const int offA = half ? lofs(k0 + 2) : lofs(k0 + 0);
        const int offB = half ? lofs(k0 + 3) : lofs(k0 + 1);

        // B-fragment: rows kA, kA+1 of W_s -- two coalesced 2x64B loads
        v2f bf;
        bf.x = wlane[(k0 + 0) * COUT];
        bf.y = wlane[(k0 + 1) * COUT];

        v2f a;
        // M-tile 0: batches 0..15
        a.x = tbase[offA] + bs;
        a.y = tbase[offB] + bs;
        acc0 = __builtin_amdgcn_wmma_f32_16x16x4_f32(false, a, false, bf,
                                                     (short)0, acc0, false, false);
        // M-tile 1: batches 16..31
        a.x = tbase[offA + 16] + bs;
        a.y = tbase[offB + 16] + bs;
        acc1 = __builtin_amdgcn_wmma_f32_16x16x4_f32(false, a, false, bf,
                                                     (short)0, acc1, false, false);
        // M-tile 2: batches 32..47
        a.x = tbase[offA + 32] + bs;
        a.y = tbase[offB + 32] + bs;
        acc2 = __builtin_amdgcn_wmma_f32_16x16x4_f32(false, a, false, bf,
                                                     (short)0, acc2, false, false);
        // M-tile 3: batches 48..63
        a.x = tbase[offA + 48] + bs;
        a.y = tbase[offB + 48] + bs;
        acc3 = __builtin_amdgcn_wmma_f32_16x16x4_f32(false, a, false, bf,
                                                     (short)0, acc3, false, false);
    }

    // ---------------- transpose results through LDS, store coalesced ----------------
    // D layout: VGPR v, lanes 0-15 -> M=v, lanes 16-31 -> M=v+8; N = lane&15.
    __syncthreads();                  // all waves done reading the X tile
    const int mbase = half * 8;
#pragma unroll
    for (int v = 0; v < 8; ++v) {
        const int m = mbase + v;
        tile[(((0  + m) * COUT + n) * NS) + wv] = acc0[v];
        tile[(((16 + m) * COUT + n) * NS) + wv] = acc1[v];
        tile[(((32 + m) * COUT + n) * NS) + wv] = acc2[v];
        tile[(((48 + m) * COUT + n) * NS) + wv] = acc3[v];
    }
    __syncthreads();

    // 1024 runs of 8 contiguous dwords: out[(b*16+c)*4096 + s0 .. s0+7]
    const int s0 = oh * WW + ow0;
#pragma unroll
    for (int p = tid; p < B_ * COUT; p += 256) {
        const float4 lo = *(const float4*)&tile[p * NS];
        const float4 hi = *(const float4*)&tile[p * NS + 4];
        float4* dst = (float4*)(out + (size_t)p * SS + s0);
        dst[0] = lo;
        dst[1] = hi;
    }
}

extern "C" void kernel_launch(void* const* d_in, const int* in_sizes, int n_in,
                              void* d_out, int out_size, void* d_ws, size_t ws_size,
                              hipStream_t stream) {
    const float* X    = (const float*)d_in[0];
    const float* Wt   = (const float*)d_in[1];
    const float* bias = (const float*)d_in[2];
    float* out        = (float*)d_out;

    const size_t smem = (size_t)TILE_FLOATS * sizeof(float); // 249600 B < 320 KB/WGP
    dim3 grid(SS / NS);   // 512 workgroups
    dim3 block(256);      // 8 wave32 waves, one per output location
    varconv_wmma_f32<<<grid, block, smem, stream>>>(X, Wt, bias, out);
}